// FullyConnectedTensorProduct_68650757259972
// MI455X (gfx1250) — compile-verified
//
#include <hip/hip_runtime.h>
#include <stdint.h>

// FullyConnectedTensorProduct, MUL=128, BATCH=1024.
// out_s[b,w]   = a0*( sum_uv W0[u,v,w]*s1[b,u]*s2[b,v] + (1/sqrt3)*sum_uv W1[u,v,w]*(v1[b,u,:].v2[b,v,:]) )
// out_v[b,w,j] = a0*( sum_uv W2[u,v,w]*s1[b,u]*v2[b,v,j] + sum_uv W3[u,v,w]*v1[b,u,j]*s2[b,v] )
// (ALPHA1*INV_SQRT3 == ALPHA0 exactly.)

#define BATCH   1024
#define ALPHA0f      0.0055242717280199f   // 1/sqrt(32768)
#define A0_ISQ3f     0.0031894377460119f   // ALPHA0/sqrt(3)

typedef float v2f __attribute__((ext_vector_type(2)));
typedef float v4f __attribute__((ext_vector_type(4)));
typedef float v8f __attribute__((ext_vector_type(8)));

// Dynamic-LDS byte layout (dynamic LDS base offset is 0: no static __shared__ here)
#define LDS_A0    0        // A tile for even stages: 16 x 128 f32 (8 KB)
#define LDS_A1    8192     // A tile for odd  stages
#define LDS_W     16384    // two W stage buffers, 128x128 f32 each (64 KB)
#define LDS_WBUF  65536
#define LDS_TOTAL 147456   // 144 KB of the WGP's 320 KB

__global__ __launch_bounds__(256) void fctp_wmma_kernel(
    const float* __restrict__ x1, const float* __restrict__ x2,
    const float* __restrict__ wgt, float* __restrict__ out)
{
  extern __shared__ char smem[];

  const int tid  = threadIdx.x;
  const int lane = tid & 31;
  const int wave = tid >> 5;          // 0..7 -> owns w-tile [wave*16, wave*16+16)
  const int c    = blockIdx.y;        // 0=scalar, 1..3=vector component j=c-1
  const int b0   = blockIdx.x * 16;   // batch tile

  // Weight blocks contracted for this channel: (t0,t1)
  const int t0 = (c == 0) ? 0 : 2;

  // ---- per-thread register cache of x2 (held for the whole kernel) ----
  // thread fills A row rfill, columns [vbase, vbase+8)
  const int rfill = tid >> 4;         // 0..15
  const int vbase = (tid & 15) * 8;   // 0..120
  float s2c[8], v2x[8], v2y[8], v2z[8];
  {
    const float* X2 = x2 + (size_t)(b0 + rfill) * 512;
#pragma unroll
    for (int q = 0; q < 8; ++q) {
      const int v = vbase + q;
      s2c[q] = X2[v];
      v2x[q] = X2[128 + 3 * v + 0];
      v2y[q] = X2[128 + 3 * v + 1];
      v2z[q] = X2[128 + 3 * v + 2];
    }
  }
  // component of v2 used by vector channels (selected once; c is block-uniform)
  float selv[8];
#pragma unroll
  for (int q = 0; q < 8; ++q) selv[q] = v2x[q];
  if (c == 2) {
#pragma unroll
    for (int q = 0; q < 8; ++q) selv[q] = v2y[q];
  } else if (c == 3) {
#pragma unroll
    for (int q = 0; q < 8; ++q) selv[q] = v2z[q];
  }

  // ---- async stage of W[t][u][:][:] (64 KB) into LDS buffer (k&1) ----
  auto issue_stage = [&](int k) {
    const int u  = k >> 1;
    const int tb = t0 + (k & 1);
    const unsigned long long gb =
        (unsigned long long)(uintptr_t)(wgt + (size_t)tb * 2097152 + (size_t)u * 16384);
    const unsigned ldsbase = (unsigned)(LDS_W + (k & 1) * LDS_WBUF);
#pragma unroll
    for (int it = 0; it < 16; ++it) {
      const unsigned off = (unsigned)((it * 256 + tid) * 16);  // 16 B per lane per issue
      const unsigned dst = ldsbase + off;
      asm volatile("global_load_async_to_lds_b128 %0, %1, %2"
                   :: "v"(dst), "v"(off), "s"(gb) : "memory");
    }
  };

  // ---- build A tiles for iteration u (A0 pairs with block t0, A1 with t0+1) ----
  auto fill_A = [&](int u) {
    const float* X1 = x1 + (size_t)(b0 + rfill) * 512;
    float a0[8], a1[8];
    if (c == 0) {
      const float s1u = ALPHA0f * X1[u];
      const float w1x = A0_ISQ3f * X1[128 + 3 * u + 0];
      const float w1y = A0_ISQ3f * X1[128 + 3 * u + 1];
      const float w1z = A0_ISQ3f * X1[128 + 3 * u + 2];
#pragma unroll
      for (int q = 0; q < 8; ++q) {
        a0[q] = s1u * s2c[q];
        a1[q] = w1x * v2x[q] + w1y * v2y[q] + w1z * v2z[q];
      }
    } else {
      const float s1u = ALPHA0f * X1[u];
      const float v1j = ALPHA0f * X1[128 + 3 * u + (c - 1)];
#pragma unroll
      for (int q = 0; q < 8; ++q) {
        a0[q] = s1u * selv[q];   // pairs with W_svv
        a1[q] = v1j * s2c[q];    // pairs with W_vsv
      }
    }
    float* A0p = (float*)(smem + LDS_A0) + rfill * 128 + vbase;
    float* A1p = (float*)(smem + LDS_A1) + rfill * 128 + vbase;
    *(v4f*)(A0p + 0) = v4f{a0[0], a0[1], a0[2], a0[3]};
    *(v4f*)(A0p + 4) = v4f{a0[4], a0[5], a0[6], a0[7]};
    *(v4f*)(A1p + 0) = v4f{a1[0], a1[1], a1[2], a1[3]};
    *(v4f*)(A1p + 4) = v4f{a1[4], a1[5], a1[6], a1[7]};
  };

  // ---- one K=128 chunk: 32 x v_wmma_f32_16x16x4_f32 ----
  const int m    = lane & 15;   // M row (A) / N col (B,C) per ISA striping
  const int half = lane >> 4;   // selects K pair {0,1} vs {2,3}
  const int wn   = wave * 16 + m;
  auto compute_stage = [&](int k, v8f& acc) {
    const float* Wl = (const float*)(smem + LDS_W + (k & 1) * LDS_WBUF);
    const float* Ap = (const float*)(smem + ((k & 1) ? LDS_A1 : LDS_A0)) + m * 128;
#pragma unroll 8
    for (int kk = 0; kk < 32; ++kk) {
      const int koff = kk * 4 + 2 * half;
      const v2f a = *(const v2f*)(Ap + koff);          // ds_load_b64
      v2f b;
      b.x = Wl[(koff + 0) * 128 + wn];                 // ds_load_b32
      b.y = Wl[(koff + 1) * 128 + wn];                 // ds_load_b32
      acc = __builtin_amdgcn_wmma_f32_16x16x4_f32(
          false, a, false, b, (short)0, acc, false, false);
    }
  };

  // ---- software-pipelined main loop: 256 stages (128 u x 2 weight blocks) ----
  v8f acc = {};
  issue_stage(0);
  for (int k = 0; k < 256; ++k) {
    if (k + 1 < 256) issue_stage(k + 1);          // overlap DMA with this stage
    if ((k & 1) == 0) fill_A(k >> 1);             // new A tiles once per u
    if (k + 1 < 256) asm volatile("s_wait_asynccnt 16" ::: "memory");
    else             asm volatile("s_wait_asynccnt 0"  ::: "memory");
    __syncthreads();                              // stage-k W + A visible to all
    compute_stage(k, acc);
    __syncthreads();                              // done before buffers reused
  }

  // ---- store 16x16 f32 accumulator (C/D striping: VGPR r -> M = r + 8*half) ----
#pragma unroll
  for (int r = 0; r < 8; ++r) {
    const int row  = b0 + r + 8 * half;
    const int wcol = wave * 16 + m;
    const int col  = (c == 0) ? wcol : (128 + 3 * wcol + (c - 1));
    out[(size_t)row * 512 + col] = acc[r];
  }
}

extern "C" void kernel_launch(void* const* d_in, const int* in_sizes, int n_in,
                              void* d_out, int out_size, void* d_ws, size_t ws_size,
                              hipStream_t stream) {
  const float* x1  = (const float*)d_in[0];
  const float* x2  = (const float*)d_in[1];
  const float* wgt = (const float*)d_in[2];
  float* out = (float*)d_out;

  (void)hipFuncSetAttribute((const void*)fctp_wmma_kernel,
                            hipFuncAttributeMaxDynamicSharedMemorySize, LDS_TOTAL);

  dim3 grid(BATCH / 16, 4, 1);   // 64 batch tiles x 4 output channels
  fctp_wmma_kernel<<<grid, 256, LDS_TOTAL, stream>>>(x1, x2, wgt, out);
}